// snn_layer_16947940950665
// MI455X (gfx1250) — compile-verified
//
#include <hip/hip_runtime.h>
#include <cstdint>
#include <cstddef>

// ---------------- LIF constants (from reference) ----------------
#define T_STEPS 1000
#define R_CONST   3000.0f
#define DT_CONST  0.01f
#define TAU_M     150.0f                 // R*C
#define DT_OVER_TM (DT_CONST / TAU_M)    // 6.6666667e-5
#define V_THR 15.0f
#define V_RST 0.0f
#define V_MAX 30.0f

// ---------------- tuning ----------------
#define BLOCK   256          // 8 wave32 per workgroup
#define VEC     4            // floats per thread (b128 path)
#define STAGES  8            // async prefetch depth (8 * 256 * 16B = 32KB LDS)

typedef __attribute__((ext_vector_type(4))) float v4f;

__device__ __forceinline__ float lif_step(float v, float i) {
    // v_euler = v + ((-v + R*i)/tau_m)*dt
    float ve = v + (R_CONST * i - v) * DT_OVER_TM;
    float vn = (v  >= V_THR) ? V_RST : ve;   // reset if previous step spiked
    return     (vn >= V_THR) ? V_MAX : vn;   // spike clamp
}

// GVS-form async copy: saddr = uniform row base (SGPR pair, advanced on SALU),
// voff = per-lane byte offset (immutable VGPR), la = LDS byte address.
__device__ __forceinline__ void async_prefetch(uint32_t la, uint32_t voff,
                                               const char* base) {
    asm volatile("global_load_async_to_lds_b128 %0, %1, %2"
                 :: "v"(la), "v"(voff), "s"(base)
                 : "memory");
}

__global__ __launch_bounds__(BLOCK) void lif_scan_kernel(
    const float* __restrict__ stim, float* __restrict__ out, int N) {

    __shared__ v4f smem[STAGES * BLOCK];

    const int tid = threadIdx.x;
    const int col = (blockIdx.x * BLOCK + tid) * VEC;
    if (col >= N) return;   // uniform per wave for our sizes

    // 32-bit LDS byte address of this lane's stage-0 slot
    // (generic shared address: low 32 bits == LDS offset on gfx1250)
    const uint32_t lds_lane = (uint32_t)(uintptr_t)(&smem[tid]);
    // per-lane byte offset within a [N]-row; constant for the whole kernel
    const uint32_t voff = (uint32_t)col * (uint32_t)sizeof(float);
    const size_t row_bytes = (size_t)N * sizeof(float);

    // ---- prologue: fill the 8-deep async pipeline (t = 0..7) ----
    const char* base = (const char*)stim;   // uniform -> SGPR pair
    #pragma unroll
    for (int s = 0; s < STAGES; ++s) {
        async_prefetch(lds_lane + (uint32_t)(s * BLOCK * sizeof(v4f)), voff, base);
        base += row_bytes;                  // scalar add, off the VALU path
    }

    float v0 = 0.f, v1 = 0.f, v2 = 0.f, v3 = 0.f;
    float* gout = out + col;

    // ---- main loop: t in [0, T-8), unrolled by STAGES so the stage
    //      index and LDS offsets are compile-time constants ----
    for (int tb = 0; tb < T_STEPS - STAGES; tb += STAGES) {
        #pragma unroll
        for (int s = 0; s < STAGES; ++s) {
            // async loads complete in order: <= STAGES-1 outstanding
            // guarantees the oldest (this stage's) load has landed in LDS.
            asm volatile("s_wait_asynccnt %0" :: "n"(STAGES - 1) : "memory");

            v4f iv = smem[s * BLOCK + tid];          // ds_load_b128
            v0 = lif_step(v0, iv.x);
            v1 = lif_step(v1, iv.y);
            v2 = lif_step(v2, iv.z);
            v3 = lif_step(v3, iv.w);

            v4f res = {v0, v1, v2, v3};
            __builtin_nontemporal_store(res, (v4f*)gout);  // b128 NT store
            gout += N;

            // prefetch timestep t+STAGES into the slot we just consumed
            // (safe: ds_load data already consumed by the VALU above)
            async_prefetch(lds_lane + (uint32_t)(s * BLOCK * sizeof(v4f)), voff, base);
            base += row_bytes;
        }
    }

    // ---- tail: last 8 timesteps, drain the pipeline ----
    asm volatile("s_wait_asynccnt 0" ::: "memory");
    #pragma unroll
    for (int s = 0; s < STAGES; ++s) {
        v4f iv = smem[s * BLOCK + tid];
        v0 = lif_step(v0, iv.x);
        v1 = lif_step(v1, iv.y);
        v2 = lif_step(v2, iv.z);
        v3 = lif_step(v3, iv.w);
        v4f res = {v0, v1, v2, v3};
        __builtin_nontemporal_store(res, (v4f*)gout);
        gout += N;
    }
}

extern "C" void kernel_launch(void* const* d_in, const int* in_sizes, int n_in,
                              void* d_out, int out_size, void* d_ws, size_t ws_size,
                              hipStream_t stream) {
    (void)n_in; (void)d_ws; (void)ws_size; (void)out_size;
    const float* stim = (const float*)d_in[0];   // [T, N] float32
    float* out = (float*)d_out;                  // [T, N] float32

    const int total = in_sizes[0];
    const int N = total / T_STEPS;               // 131072
    const int threads = N / VEC;                 // 32768
    dim3 grid((threads + BLOCK - 1) / BLOCK);    // 128 workgroups
    lif_scan_kernel<<<grid, BLOCK, 0, stream>>>(stim, out, N);
}